// GraphContrastiveLoss_26010321944908
// MI455X (gfx1250) — compile-verified
//
#include <hip/hip_runtime.h>
#include <math.h>

// GraphContrastiveLoss on MI455X (gfx1250, wave32).
//
// Design: latency-bound problem (192KB gathered, ~10ns of HBM time at
// 23.3TB/s). One 640-thread block = 20 wave32s on one WGP. Each wave owns one
// 16x16 Gram tile (5 tiles per group of 16 samples x 4 groups) and computes it
// as a chain of 64 V_WMMA_F32_16X16X4_F32 ops (fp32 WMMA -> bit-accurate vs
// the fp32 reference, unlike f16/bf16 paths). 4 interleaved accumulators
// break the serial D->C dependency of the WMMA chain.

typedef __attribute__((ext_vector_type(2))) float v2f;
typedef __attribute__((ext_vector_type(8))) float v8f;

#define B_  64
#define N_  4096
#define D_  256
#define INV_TEMP 10.0f   // 1 / 0.1
#define EPS_ 1e-8f

__global__ __launch_bounds__(640) void GraphContrastiveLoss_kernel(
    const float* __restrict__ emb,        // (B, N, D) fp32
    const int*   __restrict__ anchor_idx, // (B,)
    const int*   __restrict__ pos_idx,    // (B,)
    const int*   __restrict__ neg_gidx,   // (B,)
    const int*   __restrict__ neg_nidx,   // (B,)
    float*       __restrict__ out)        // scalar
{
    __shared__ float sims[5][B_];     // [pair][sample]: aa, pp, nn, ap, an
    __shared__ float per_sample[B_];

    const int tid  = threadIdx.x;
    const int wave = tid >> 5;        // 0..19
    const int lane = tid & 31;
    const int g    = wave / 5;        // sample group 0..3 (16 samples each)
    const int q    = wave % 5;        // 0:aa 1:pp 2:nn 3:ap 4:an
    const int row  = lane & 15;       // tile row/col handled by this lane
    const int h    = lane >> 4;       // lane half selects K pair {2h, 2h+1}
    const int s    = g * 16 + row;    // sample index

    // Gather the three row base pointers for this sample.
    const float* arow = emb + ((size_t)s * N_ + (size_t)anchor_idx[s]) * D_;
    const float* prow = emb + ((size_t)s * N_ + (size_t)pos_idx[s])    * D_;
    const float* nrow = emb + ((size_t)neg_gidx[s] * N_ + (size_t)neg_nidx[s]) * D_;

    const float* xrow;
    const float* yrow;
    switch (q) {                       // wave-uniform: no EXEC divergence
        case 0:  xrow = arow; yrow = arow; break;
        case 1:  xrow = prow; yrow = prow; break;
        case 2:  xrow = nrow; yrow = nrow; break;
        case 3:  xrow = arow; yrow = prow; break;
        default: xrow = arow; yrow = nrow; break;
    }

    // 16x16x4 f32 fragments: A(16x4) and B(4x16)=Y^T share the per-lane map
    //   frag[v] = Row[lane&15][kbase + 2*(lane>>4) + v]
    const float* xp = xrow + 2 * h;
    const float* yp = yrow + 2 * h;

    v8f acc0 = {}, acc1 = {}, acc2 = {}, acc3 = {};
#pragma unroll
    for (int kk = 0; kk < D_ / 16; ++kk) {   // 16 iters x 4 WMMA (K=4 each)
        const float* x = xp + kk * 16;
        const float* y = yp + kk * 16;
        v2f a0 = *(const v2f*)(x + 0);
        v2f b0 = *(const v2f*)(y + 0);
        v2f a1 = *(const v2f*)(x + 4);
        v2f b1 = *(const v2f*)(y + 4);
        v2f a2 = *(const v2f*)(x + 8);
        v2f b2 = *(const v2f*)(y + 8);
        v2f a3 = *(const v2f*)(x + 12);
        v2f b3 = *(const v2f*)(y + 12);
        acc0 = __builtin_amdgcn_wmma_f32_16x16x4_f32(false, a0, false, b0,
                                                     (short)0, acc0, false, false);
        acc1 = __builtin_amdgcn_wmma_f32_16x16x4_f32(false, a1, false, b1,
                                                     (short)0, acc1, false, false);
        acc2 = __builtin_amdgcn_wmma_f32_16x16x4_f32(false, a2, false, b2,
                                                     (short)0, acc2, false, false);
        acc3 = __builtin_amdgcn_wmma_f32_16x16x4_f32(false, a3, false, b3,
                                                     (short)0, acc3, false, false);
    }
    v8f acc = acc0 + acc1 + acc2 + acc3;

    // Diagonal extraction (f32 C/D layout):
    //   diag(i), i<8 : VGPR i,   lane i
    //   diag(i), i>=8: VGPR i-8, lane i+16
    const int vsel = lane & 7;
    float d = acc[0];
    d = (vsel == 1) ? acc[1] : d;
    d = (vsel == 2) ? acc[2] : d;
    d = (vsel == 3) ? acc[3] : d;
    d = (vsel == 4) ? acc[4] : d;
    d = (vsel == 5) ? acc[5] : d;
    d = (vsel == 6) ? acc[6] : d;
    d = (vsel == 7) ? acc[7] : d;

    const bool writer = (lane < 8) || (lane >= 24);
    const int  di     = (lane < 8) ? lane : (lane - 16);
    if (writer) sims[q][g * 16 + di] = d;

    __syncthreads();

    if (tid < B_) {
        const float aa = sims[0][tid];
        const float pp = sims[1][tid];
        const float nn = sims[2][tid];
        const float ap = sims[3][tid];
        const float an = sims[4][tid];
        // torch cosine_similarity eps semantics: dot / (max(|a|,eps)*max(|b|,eps))
        const float na  = fmaxf(sqrtf(aa), EPS_);
        const float np  = fmaxf(sqrtf(pp), EPS_);
        const float nng = fmaxf(sqrtf(nn), EPS_);
        const float pos = (ap / (na * np))  * INV_TEMP;
        const float neg = (an / (na * nng)) * INV_TEMP;
        // logaddexp(0, x) = max(x,0) + log1p(exp(-|x|))
        const float x = neg - pos;
        per_sample[tid] = fmaxf(x, 0.0f) + log1pf(expf(-fabsf(x)));
    }

    __syncthreads();

    if (tid == 0) {
        float sum = 0.0f;
#pragma unroll
        for (int i = 0; i < B_; ++i) sum += per_sample[i];
        out[0] = sum * (1.0f / (float)B_);
    }
}

extern "C" void kernel_launch(void* const* d_in, const int* in_sizes, int n_in,
                              void* d_out, int out_size, void* d_ws, size_t ws_size,
                              hipStream_t stream) {
    (void)in_sizes; (void)n_in; (void)d_ws; (void)ws_size; (void)out_size;
    const float* emb        = (const float*)d_in[0];
    // d_in[1] = graph_labels: unused by the reference computation
    const int*   anchor_idx = (const int*)d_in[2];
    const int*   pos_idx    = (const int*)d_in[3];
    const int*   neg_gidx   = (const int*)d_in[4];
    const int*   neg_nidx   = (const int*)d_in[5];
    float*       out        = (float*)d_out;

    GraphContrastiveLoss_kernel<<<1, 640, 0, stream>>>(
        emb, anchor_idx, pos_idx, neg_gidx, neg_nidx, out);
}